// W2VBert2Processor_39857296507433
// MI455X (gfx1250) — compile-verified
//
#include <hip/hip_runtime.h>

// ---------------------------------------------------------------------------
// W2VBert2 feature extractor for MI455X (gfx1250, wave32, WMMA + TDM).
//
// Linear front end (mean removal, pre-emphasis, window, 2^15 scale) folded
// into combined 400x257 DFT re/im matrices -> rFFT power spectrum and mel
// projection are both f32 WMMA GEMMs (16x16x4).
//
// Data movement uses the Tensor Data Mover:
//  - raw audio:   1-D TDM load with pad_interval=64dw/pad_amount=1dw, which
//                 implements the a+(a>>6) LDS bank swizzle in hardware and
//                 zero-fills past end-of-utterance via tensor_dim0.
//  - basis/mel:   2-D strided TDM tile loads (row pitch via dim0_stride).
//  - log-mel out: TDM store from LDS; tensor_dim1=fvalid drops OOB rows of
//                 the partial last frame group.
// Roofline: ~155 MB HBM traffic -> ~7 us floor @ 23.3 TB/s; ~35 GFLOP of
// f32 WMMA is negligible. Basis (870 KB) + mel (87 KB) stay L2-resident.
// ---------------------------------------------------------------------------

#define B_BATCH   32
#define N_SAMP    480000
#define FRAME_LEN 400
#define HOP       160
#define NBINS     257
#define KPAD      272            // 17 * 16 (freq bins padded for WMMA tiles)
#define NMEL      80             // 5 * 16
#define FPB       2998           // frames per batch: 1 + (480000-400)/160
#define NF_OUT    1499           // FPB / 2
#define T_OUT     1500           // NF_OUT + 1 pad frame
#define FPG       128            // frames per workgroup (8 waves x 16)
#define GPB       24             // ceil(FPB / FPG)
#define XS_RAW    20720          // (FPG-1)*HOP + FRAME_LEN samples staged
#define XS_ALLOC  21056          // swizzled allocation (>= XS_RAW + XS_RAW/64)
#define SPW_STRIDE 274           // LDS power-row stride: 274 mod 64 = 18 ->
                                 // distinct banks across 16 lanes
#define MEL_FLOOR 1.192092955078125e-07f

typedef float v2f __attribute__((ext_vector_type(2)));
typedef float v8f __attribute__((ext_vector_type(8)));
typedef unsigned int u32x4 __attribute__((ext_vector_type(4)));
typedef int i32x4 __attribute__((ext_vector_type(4)));
typedef int i32x8 __attribute__((ext_vector_type(8)));

// LDS bank swizzle for hop-160 frame rows (160 % 64 == 32 would be 8-way
// conflicted). Matches TDM pad: 1 dword inserted after every 64 dwords.
__device__ __forceinline__ int xsw(int a) { return a + (a >> 6); }

// ---------------------------------------------------------------------------
// TDM descriptor packing (ISA 8.3/8.4): 2-D tile, 4-byte elements.
// group0: [1:0]=count=1 | [63:32]=lds_addr | [120:64]=global_addr | type=2
// group1: data_size=2(4B), optional pad(64dw interval, 1dw amount),
//         tensor_dim0/1, tile_dim0/1, tensor_dim0_stride.
// ---------------------------------------------------------------------------
__device__ __forceinline__ void tdm_desc(u32x4& g0, i32x8& g1,
    unsigned lds_byte_off, const void* gptr, unsigned tile_w, unsigned tile_h,
    unsigned tensor_w, unsigned tensor_h, unsigned stride_elems, bool pad64) {
  unsigned long long ga = (unsigned long long)gptr;
  g0.x = 1u;                                  // count=1, user descriptor
  g0.y = lds_byte_off;                        // lds_addr (bytes)
  g0.z = (unsigned)ga;                        // global_addr[31:0]
  g0.w = (unsigned)(ga >> 32) | 0x80000000u;  // global_addr[56:32] | type=2
  int w0 = (2 << 16);                         // data_size = 4 bytes
  if (pad64) w0 |= (1 << 20) | (5 << 22);     // pad_enable, 64dw interval, 1dw
  g1[0] = w0;
  g1[1] = (int)((tensor_w & 0xFFFFu) << 16);                    // dim0[15:0]
  g1[2] = (int)((tensor_w >> 16) | ((tensor_h & 0xFFFFu) << 16));
  g1[3] = (int)((tensor_h >> 16) | ((tile_w & 0xFFFFu) << 16)); // tile_dim0
  g1[4] = (int)(tile_h & 0xFFFFu);                              // tile_dim1
  g1[5] = (int)stride_elems;                  // tensor_dim0_stride[31:0]
  g1[6] = 0;
  g1[7] = 0;
}

__device__ __forceinline__ void tdm_load2d(unsigned lds_byte_off,
    const void* g, unsigned tw, unsigned th, unsigned sw, unsigned sh,
    unsigned stride, bool pad64) {
  u32x4 g0; i32x8 g1;
  tdm_desc(g0, g1, lds_byte_off, g, tw, th, sw, sh, stride, pad64);
  i32x4 z4 = {0, 0, 0, 0};
  i32x8 z8 = {0, 0, 0, 0, 0, 0, 0, 0};
  __builtin_amdgcn_tensor_load_to_lds(g0, g1, z4, z4, z8, 0);
}

__device__ __forceinline__ void tdm_store2d(unsigned lds_byte_off,
    void* g, unsigned tw, unsigned th, unsigned sw, unsigned sh,
    unsigned stride) {
  u32x4 g0; i32x8 g1;
  tdm_desc(g0, g1, lds_byte_off, g, tw, th, sw, sh, stride, false);
  i32x4 z4 = {0, 0, 0, 0};
  i32x8 z8 = {0, 0, 0, 0, 0, 0, 0, 0};
  __builtin_amdgcn_tensor_store_from_lds(g0, g1, z4, z4, z8, 0);
}

// ---------------------------------------------------------------------------
// Kernel 0a: combined (window * preemph * mean-removal * 2^15 * DFT) basis.
// B'[s][k] = 32768 * (A[s] - (sum_t A[t]) / 400), where
// A[t] = c0(t)*w[t]*trig(t,k) - 0.97*w[t+1]*trig(t+1,k),  c0(0)=0.03.
// ---------------------------------------------------------------------------
__global__ __launch_bounds__(512) void build_basis_kernel(
    const float* __restrict__ window, float* __restrict__ bRe,
    float* __restrict__ bIm) {
  __shared__ float wc[401], ws_[401];
  __shared__ float red[512];
  const int k = blockIdx.x;     // 0..KPAD-1
  const int t = threadIdx.x;    // 0..511

  if (t < 400) {
    int ph = (t * k) & 511;     // exact phase mod 512
    float ang = (float)ph * 0.01227184630308513f;   // 2*pi/512
    float w = window[t];
    wc[t]  = w * __cosf(ang);
    ws_[t] = w * __sinf(ang);
  } else if (t == 400) {
    wc[400] = 0.0f; ws_[400] = 0.0f;
  }
  __syncthreads();

  float Ac = 0.0f, As = 0.0f;
  if (t < 400) {
    float c0 = (t == 0) ? 0.03f : 1.0f;
    Ac = c0 * wc[t]  - 0.97f * wc[t + 1];
    As = c0 * ws_[t] - 0.97f * ws_[t + 1];
  }

  red[t] = Ac;
  __syncthreads();
  for (int s = 256; s > 0; s >>= 1) {
    if (t < s) red[t] += red[t + s];
    __syncthreads();
  }
  float Sc = red[0];
  __syncthreads();
  red[t] = As;
  __syncthreads();
  for (int s = 256; s > 0; s >>= 1) {
    if (t < s) red[t] += red[t + s];
    __syncthreads();
  }
  float Ss = red[0];

  if (t < 400) {
    float re = 0.0f, im = 0.0f;
    if (k < NBINS) {
      re = 32768.0f * (Ac - Sc * (1.0f / 400.0f));
      im = 32768.0f * (As - Ss * (1.0f / 400.0f));
    }
    bRe[t * KPAD + k] = re;
    bIm[t * KPAD + k] = im;
  }
}

// ---------------------------------------------------------------------------
// Kernel 0b: zero-pad mel filters to [KPAD][80].
// ---------------------------------------------------------------------------
__global__ __launch_bounds__(256) void build_mel_kernel(
    const float* __restrict__ mel, float* __restrict__ melPad) {
  int i = blockIdx.x * 256 + threadIdx.x;
  if (i < KPAD * NMEL) {
    int kk = i / NMEL;
    melPad[i] = (kk < NBINS) ? mel[i] : 0.0f;
  }
}

// ---------------------------------------------------------------------------
// Kernel 1: fused framing + DFT power + mel + log.
// Grid: 32 batches * 24 groups; block: 256 threads (8 wave32s).
// Wave 0 drives all TDM traffic; every wave owns one 16-frame row tile.
// LDS: xs 21056 | sbR 6400 | sbI 6400 | spw 128*274 | smel 128*80  floats
//      = 316,672 bytes (< 320 KB/WGP).
// ---------------------------------------------------------------------------
__global__ __launch_bounds__(256) void spectro_kernel(
    const float* __restrict__ raw, const float* __restrict__ bRe,
    const float* __restrict__ bIm, const float* __restrict__ melPad,
    float* __restrict__ feats) {
  extern __shared__ float sm[];
  float* xs   = sm;                      // swizzled raw audio (TDM pad)
  float* sbR  = sm + XS_ALLOC;           // 400*16 basis-re tile
  float* sbI  = sbR + 400 * 16;          // 400*16 basis-im tile
  float* spw  = sbI + 400 * 16;          // 128 * SPW_STRIDE power
  float* smel = spw + FPG * SPW_STRIDE;  // 128 * 80 log-mel staging

  // LDS aperture: low 32 bits of a generic LDS address == LDS byte offset.
  const unsigned lds0     = (unsigned)(unsigned long long)(void*)sm;
  const unsigned off_sbR  = lds0 + XS_ALLOC * 4u;
  const unsigned off_sbI  = off_sbR + 400 * 16 * 4u;
  const unsigned off_smel = off_sbI + (400 * 16 + FPG * SPW_STRIDE) * 4u;

  const int b  = blockIdx.x / GPB;
  const int g  = blockIdx.x % GPB;
  const int f0 = g * FPG;
  const float* rb = raw + (size_t)b * N_SAMP;
  const int base = f0 * HOP;
  int fvalid = FPB - f0;
  if (fvalid > FPG) fvalid = FPG;

  const bool leader = (threadIdx.x < 32);   // wave 0 issues TDM

  // Raw audio: one 1-D TDM load; hardware pad = bank swizzle; OOB zero-fill.
  if (leader) {
    tdm_load2d(lds0, rb + base, XS_RAW, 1,
               (unsigned)(N_SAMP - base), 1, XS_RAW, true);
  }

  const int lane  = threadIdx.x & 31;   // wave32
  const int wave  = threadIdx.x >> 5;
  const int mrow  = lane & 15;          // M (and N) index within tile
  const int khalf = lane >> 4;          // which K pair this lane carries
  const int arow_base = (wave * 16 + mrow) * HOP;

  // ---- DFT: 17 freq tiles, K = 400 in steps of 4 ----
  for (int n = 0; n < KPAD / 16; ++n) {
    __syncthreads();  // prior-iteration readers done with sbR/sbI
    if (leader) {     // stage basis tile [400][16] via strided TDM loads
      tdm_load2d(off_sbR, bRe + n * 16, 16, 400, 16, 400, KPAD, false);
      tdm_load2d(off_sbI, bIm + n * 16, 16, 400, 16, 400, KPAD, false);
      __builtin_amdgcn_s_wait_tensorcnt(0);   // also covers the xs load
    }
    __syncthreads();

    v8f accR = {0.f, 0.f, 0.f, 0.f, 0.f, 0.f, 0.f, 0.f};
    v8f accI = {0.f, 0.f, 0.f, 0.f, 0.f, 0.f, 0.f, 0.f};
    for (int k0 = 0; k0 < FRAME_LEN; k0 += 4) {
      int ka = k0 + 2 * khalf;
      v2f a, br, bi;
      a.x  = xs[xsw(arow_base + ka)];
      a.y  = xs[xsw(arow_base + ka + 1)];
      br.x = sbR[ka * 16 + mrow];
      br.y = sbR[(ka + 1) * 16 + mrow];
      bi.x = sbI[ka * 16 + mrow];
      bi.y = sbI[(ka + 1) * 16 + mrow];
      accR = __builtin_amdgcn_wmma_f32_16x16x4_f32(
          false, a, false, br, (short)0, accR, false, false);
      accI = __builtin_amdgcn_wmma_f32_16x16x4_f32(
          false, a, false, bi, (short)0, accI, false, false);
    }
    __syncthreads();  // all waves finished reading sbR/sbI for this tile

    // Power spectrum -> LDS (wave writes only its own 16 rows).
#pragma unroll
    for (int j = 0; j < 8; ++j) {
      int fr = wave * 16 + j + 8 * khalf;    // D row: M = j + 8*(lane/16)
      float re = accR[j], im = accI[j];
      spw[fr * SPW_STRIDE + n * 16 + mrow] = re * re + im * im;
    }
  }

  // ---- Mel projection: K = 272 in steps of 4, 5 output tiles of 16 ----
  const int sprow = (wave * 16 + mrow) * SPW_STRIDE;
  for (int m = 0; m < NMEL / 16; ++m) {
    __syncthreads();  // previous users of sbR done (incl. last DFT tile)
    if (leader) {     // stage mel tile [272][16] via strided TDM load
      tdm_load2d(off_sbR, melPad + m * 16, 16, KPAD, 16, KPAD, NMEL, false);
      __builtin_amdgcn_s_wait_tensorcnt(0);
    }
    __syncthreads();

    v8f acc = {0.f, 0.f, 0.f, 0.f, 0.f, 0.f, 0.f, 0.f};
    for (int k0 = 0; k0 < KPAD; k0 += 4) {
      int ka = k0 + 2 * khalf;
      v2f a, bm;
      a.x  = spw[sprow + ka];
      a.y  = spw[sprow + ka + 1];
      bm.x = sbR[ka * 16 + mrow];
      bm.y = sbR[(ka + 1) * 16 + mrow];
      acc = __builtin_amdgcn_wmma_f32_16x16x4_f32(
          false, a, false, bm, (short)0, acc, false, false);
    }
#pragma unroll
    for (int j = 0; j < 8; ++j) {
      int fr = wave * 16 + j + 8 * khalf;
      float v = fmaxf(acc[j], MEL_FLOOR);
      smel[fr * NMEL + m * 16 + mrow] = __logf(v);  // conflict-free rows
    }
  }
  __syncthreads();

  // One TDM store per block; tensor_dim1 = fvalid drops OOB frame rows.
  if (leader) {
    tdm_store2d(off_smel, feats + ((size_t)b * FPB + f0) * NMEL,
                NMEL, FPG, NMEL, (unsigned)fvalid, NMEL);
    __builtin_amdgcn_s_wait_tensorcnt(0);  // LDS must stay live until done
  }
}

// ---------------------------------------------------------------------------
// Kernel 2: per-batch mean/var (ddof=1) over 2998 frames, normalize,
// stack frame pairs, pad row of 1.0, write int32 attention mask.
// ---------------------------------------------------------------------------
__global__ __launch_bounds__(256) void norm_emit_kernel(
    const float* __restrict__ feats, float* __restrict__ outF,
    int* __restrict__ outM) {
  __shared__ float ps[240], pq[240];
  __shared__ float smean[NMEL], srstd[NMEL];
  const int b = blockIdx.x;
  const int t = threadIdx.x;
  const float* fb = feats + (size_t)b * FPB * NMEL;

  if (t < 240) {          // 3 partials x 80 mels, fully coalesced reads
    int m = t % NMEL, part = t / NMEL;
    float s = 0.0f, q = 0.0f;
    for (int f = part; f < FPB; f += 3) {
      float v = fb[f * NMEL + m];
      s += v; q += v * v;
    }
    ps[t] = s; pq[t] = q;
  }
  __syncthreads();
  if (t < NMEL) {
    float S = ps[t] + ps[t + 80] + ps[t + 160];
    float Q = pq[t] + pq[t + 80] + pq[t + 160];
    float mean = S * (1.0f / FPB);
    float var  = (Q - S * mean) * (1.0f / (FPB - 1));   // ddof = 1
    smean[t] = mean;
    srstd[t] = rsqrtf(var + 1e-7f);
  }
  __syncthreads();

  float* ob = outF + (size_t)b * T_OUT * 160;
  const int NE = NF_OUT * 160;
  for (int i = t; i < NE; i += 256) {
    int tt = i / 160, c = i - tt * 160;
    int half = (c >= 80) ? 1 : 0;
    int m = c - 80 * half;
    float v = fb[(2 * tt + half) * NMEL + m];
    ob[i] = (v - smean[m]) * srstd[m];
  }
  if (t < 160) ob[NF_OUT * 160 + t] = 1.0f;            // PAD_VALUE row
  for (int i = t; i < T_OUT; i += 256)
    outM[b * T_OUT + i] = (i < NF_OUT) ? 1 : 0;        // int32 mask
}

// ---------------------------------------------------------------------------
// Host-side launcher.  Inputs: [0] raw_speech f32 (32,480000),
// [1] window f32 (400), [2] mel_filters f32 (257,80).
// Output: features f32 (32,1500,160) then mask i32 (32,1500), flat.
// Workspace (floats): bRe[400*272] | bIm[400*272] | melPad[272*80]
//                   | feats[32*2998*80]  (~31.7 MB total)
// ---------------------------------------------------------------------------
extern "C" void kernel_launch(void* const* d_in, const int* in_sizes, int n_in,
                              void* d_out, int out_size, void* d_ws,
                              size_t ws_size, hipStream_t stream) {
  (void)in_sizes; (void)n_in; (void)out_size; (void)ws_size;
  const float* raw    = (const float*)d_in[0];
  const float* window = (const float*)d_in[1];
  const float* mel    = (const float*)d_in[2];

  float* ws      = (float*)d_ws;
  float* bRe     = ws;                               // 108800
  float* bIm     = bRe + 400 * KPAD;                 // 108800
  float* melPad  = bIm + 400 * KPAD;                 // 21760
  float* feats   = melPad + KPAD * NMEL;             // 32*2998*80

  float* outF = (float*)d_out;
  int*   outM = (int*)((float*)d_out + (size_t)B_BATCH * T_OUT * 160);

  build_basis_kernel<<<KPAD, 512, 0, stream>>>(window, bRe, bIm);
  build_mel_kernel<<<(KPAD * NMEL + 255) / 256, 256, 0, stream>>>(mel, melPad);

  size_t smem = (size_t)(XS_ALLOC + 2 * (400 * 16) + FPG * SPW_STRIDE +
                         FPG * NMEL) * sizeof(float);   // 316,672 B
  spectro_kernel<<<B_BATCH * GPB, 256, smem, stream>>>(raw, bRe, bIm, melPad,
                                                       feats);
  norm_emit_kernel<<<B_BATCH, 256, 0, stream>>>(feats, outF, outM);
}